// SemanticBasis_34282428957309
// MI455X (gfx1250) — compile-verified
//
#include <hip/hip_runtime.h>
#include <math.h>

// Problem shapes (match reference harness)
constexpr int V_  = 50257;
constexpr int K_  = 1024;
constexpr int D_  = 1024;
constexpr int RH_ = 384;
constexpr int RC_ = 32;
constexpr int BT_ = 4096;        // B*T
constexpr int RF_ = RH_ + RC_;   // 416 (== 26*16)

constexpr int CT_PER_WG_ = 8;    // k_cold: column tiles swept per workgroup (128 cols)

typedef __attribute__((ext_vector_type(2))) float v2f;
typedef __attribute__((ext_vector_type(8))) float v8f;

__device__ __forceinline__ v8f wmma_f32_k4(v2f a, v2f b, v8f c) {
    // D = A(16x4) * B(4x16) + C   — v_wmma_f32_16x16x4_f32
    return __builtin_amdgcn_wmma_f32_16x16x4_f32(false, a, false, b, (short)0, c,
                                                 false, false);
}

// ---------------------------------------------------------------------------
// K_inv: hot_pos[j] = v such that old_to_new[v] == j (for j < K)
// ---------------------------------------------------------------------------
__global__ __launch_bounds__(256) void k_invmap(const int* __restrict__ o2n,
                                                int* __restrict__ hot_pos) {
    int v = blockIdx.x * 256 + threadIdx.x;
    if (v >= V_) return;
    int j = o2n[v];
    if (j < K_) hot_pos[j] = v;
}

// ---------------------------------------------------------------------------
// K1: masked-gather A_full [BT, 416]; zeros implement the hot/cold select
// ---------------------------------------------------------------------------
__global__ __launch_bounds__(256) void k_build_a(const int* __restrict__ tokens,
                                                 const int* __restrict__ o2n,
                                                 const float* __restrict__ Uh,
                                                 const float* __restrict__ Uc,
                                                 float* __restrict__ A) {
    int t = blockIdx.y;
    int c = blockIdx.x * 256 + threadIdx.x;
    if (c >= RF_) return;
    int j = o2n[tokens[t]];
    bool hot = j < K_;
    float v;
    if (c < RH_) v = hot ? Uh[j * RH_ + c] : 0.0f;
    else         v = hot ? 0.0f : Uc[(j - K_) * RC_ + (c - RH_)];
    A[t * RF_ + c] = v;
}

// ---------------------------------------------------------------------------
// K0: Gram matrix G = B_full @ B_full^T   (416x416, contraction over D=1024)
// One wave per 16x16 tile, 256 WMMA steps.
// ---------------------------------------------------------------------------
__global__ __launch_bounds__(32) void k_gram(const float* __restrict__ Bh,
                                             const float* __restrict__ Bc,
                                             float* __restrict__ G) {
    int tile = blockIdx.x;                 // 0 .. 26*26-1
    int it = tile / 26, jt = tile % 26;
    int lane = threadIdx.x & 31;
    int half = lane >> 4;
    int l16  = lane & 15;
    int i = it * 16 + l16;
    int j = jt * 16 + l16;
    const float* rowA = (i < RH_) ? (Bh + (size_t)i * D_) : (Bc + (size_t)(i - RH_) * D_);
    const float* rowB = (j < RH_) ? (Bh + (size_t)j * D_) : (Bc + (size_t)(j - RH_) * D_);
    v8f c = {0.f, 0.f, 0.f, 0.f, 0.f, 0.f, 0.f, 0.f};
    for (int k = 0; k < D_; k += 4) {
        v2f a, b;
        a.x = rowA[k + half * 2 + 0];
        a.y = rowA[k + half * 2 + 1];
        b.x = rowB[k + half * 2 + 0];   // B^T[k, n] = B_full[j_n, k]
        b.y = rowB[k + half * 2 + 1];
        c = wmma_f32_k4(a, b, c);
    }
#pragma unroll
    for (int r = 0; r < 8; ++r) {
        int m = it * 16 + r + half * 8;
        G[m * RF_ + jt * 16 + l16] = c[r];
    }
}

// ---------------------------------------------------------------------------
// K2: P = A_full @ G   [BT, 416] — replaces latent formation + re-projection
// 8 waves per WG: 128 rows x 16 cols; G column tile staged in LDS.
// ---------------------------------------------------------------------------
__global__ __launch_bounds__(256) void k_project(const float* __restrict__ A,
                                                 const float* __restrict__ G,
                                                 float* __restrict__ P) {
    __shared__ float sg[16 * 417];         // [n][k], padded stride
    int nBase = blockIdx.x * 16;
    for (int e = threadIdx.x; e < 16 * RF_; e += 256) {
        int n = e / RF_;
        int k = e - n * RF_;
        sg[n * 417 + k] = G[k * RF_ + nBase + n];
    }
    __syncthreads();

    int lane = threadIdx.x & 31;
    int half = lane >> 4;
    int l16  = lane & 15;
    int w    = threadIdx.x >> 5;
    int mBase = blockIdx.y * 128 + w * 16;
    const float* arow = A + (size_t)(mBase + l16) * RF_;
    const float* bcol = sg + l16 * 417;

    v8f c = {0.f, 0.f, 0.f, 0.f, 0.f, 0.f, 0.f, 0.f};
#pragma unroll 4
    for (int k = 0; k < RF_; k += 4) {
        v2f a, b;
        a.x = arow[k + half * 2 + 0];
        a.y = arow[k + half * 2 + 1];
        b.x = bcol[k + half * 2 + 0];
        b.y = bcol[k + half * 2 + 1];
        c = wmma_f32_k4(a, b, c);
    }
#pragma unroll
    for (int r = 0; r < 8; ++r)
        P[(size_t)(mBase + r + half * 8) * RF_ + nBase + l16] = c[r];
}

// ---------------------------------------------------------------------------
// K3 (dominant): cold logits for ALL V columns, with the old_to_new gather,
// alpha scale and bias fused into the GEMM epilogue. Hot columns get a zero
// B-column here (write = bias) and are overwritten by k_hot afterwards.
//
// Each WG: 128 rows x 128 columns. The per-wave A fragments (16 rows x K=32
// of P_cold = 8 x v2f registers) are loaded ONCE and reused across 8 column
// tiles, cutting P re-reads from L2 by 8x. The gathered U_cold tile is
// double-buffered in LDS so only one barrier per tile is needed.
// ---------------------------------------------------------------------------
__global__ __launch_bounds__(256) void k_cold(const float* __restrict__ P,
                                              const float* __restrict__ Uc,
                                              const int* __restrict__ o2n,
                                              const float* __restrict__ bias,
                                              const float* __restrict__ log_alpha,
                                              float* __restrict__ out) {
    __shared__ float su[2][16 * 33];       // gathered U_cold cols [n][k], padded

    int lane = threadIdx.x & 31;
    int half = lane >> 4;
    int l16  = lane & 15;
    int w    = threadIdx.x >> 5;
    int mBase = blockIdx.y * 128 + w * 16;
    int colBase0 = blockIdx.x * (16 * CT_PER_WG_);

    // Persistent A fragments: P_cold rows for this wave's 16 output rows.
    const float* arow = P + (size_t)(mBase + l16) * RF_ + RH_;
    v2f afrag[RC_ / 4];
#pragma unroll
    for (int kk = 0; kk < RC_ / 4; ++kk) {
        afrag[kk].x = arow[kk * 4 + half * 2 + 0];
        afrag[kk].y = arow[kk * 4 + half * 2 + 1];
    }
    float alpha = __expf(log_alpha[0]);

    for (int ct = 0; ct < CT_PER_WG_; ++ct) {
        int colBase = colBase0 + ct * 16;
        float* buf = su[ct & 1];

        // Gather 16 U_cold columns (zero for hot / out-of-range columns).
        for (int e = threadIdx.x; e < 16 * 32; e += 256) {
            int cc = e >> 5, k = e & 31;
            int v = colBase + cc;
            float val = 0.0f;
            if (v < V_) {
                int j = o2n[v];                    // scalar-broadcast per 32 threads
                if (j >= K_) val = Uc[(size_t)(j - K_) * RC_ + k];
            }
            buf[cc * 33 + k] = val;
        }
        __syncthreads();   // double buffer: next iter writes the other buffer

        const float* bcol = buf + l16 * 33;
        v8f c = {0.f, 0.f, 0.f, 0.f, 0.f, 0.f, 0.f, 0.f};
#pragma unroll
        for (int kk = 0; kk < RC_ / 4; ++kk) {
            v2f b;
            b.x = bcol[kk * 4 + half * 2 + 0];
            b.y = bcol[kk * 4 + half * 2 + 1];
            c = wmma_f32_k4(afrag[kk], b, c);
        }

        int col = colBase + l16;
        if (col < V_) {
            float bv = bias[col];
#pragma unroll
            for (int r = 0; r < 8; ++r) {
                int m = mBase + r + half * 8;
                out[(size_t)m * V_ + col] = alpha * c[r] + bv;
            }
        }
    }
}

// ---------------------------------------------------------------------------
// K4: hot logits (1024 columns, rank-384), scatter-overwrite via hot_pos.
// Runs after k_cold on the same stream.
// ---------------------------------------------------------------------------
__global__ __launch_bounds__(256) void k_hot(const float* __restrict__ P,
                                             const float* __restrict__ Uh,
                                             const int* __restrict__ hot_pos,
                                             const float* __restrict__ bias,
                                             const float* __restrict__ log_alpha,
                                             float* __restrict__ out) {
    __shared__ float su[16 * 385];         // U_hot col tile [n][k], padded
    int colBase = blockIdx.x * 16;         // hot column id base (< 1024)
    for (int e = threadIdx.x; e < 16 * RH_; e += 256) {
        int cc = e / RH_;
        int k  = e - cc * RH_;
        su[cc * 385 + k] = Uh[(size_t)(colBase + cc) * RH_ + k];
    }
    __syncthreads();

    int lane = threadIdx.x & 31;
    int half = lane >> 4;
    int l16  = lane & 15;
    int w    = threadIdx.x >> 5;
    int mBase = blockIdx.y * 128 + w * 16;
    const float* arow = P + (size_t)(mBase + l16) * RF_;         // P_hot part
    const float* bcol = su + l16 * 385;

    v8f c = {0.f, 0.f, 0.f, 0.f, 0.f, 0.f, 0.f, 0.f};
#pragma unroll 4
    for (int k = 0; k < RH_; k += 4) {
        v2f a, b;
        a.x = arow[k + half * 2 + 0];
        a.y = arow[k + half * 2 + 1];
        b.x = bcol[k + half * 2 + 0];
        b.y = bcol[k + half * 2 + 1];
        c = wmma_f32_k4(a, b, c);
    }

    float alpha = __expf(log_alpha[0]);
    int j = colBase + l16;
    int v = hot_pos[j];
    float bv = bias[v];
#pragma unroll
    for (int r = 0; r < 8; ++r) {
        int m = mBase + r + half * 8;
        out[(size_t)m * V_ + v] = alpha * c[r] + bv;
    }
}

// ---------------------------------------------------------------------------
extern "C" void kernel_launch(void* const* d_in, const int* in_sizes, int n_in,
                              void* d_out, int out_size, void* d_ws, size_t ws_size,
                              hipStream_t stream) {
    (void)in_sizes; (void)n_in; (void)out_size; (void)ws_size;
    const int*   tokens = (const int*)d_in[0];
    const int*   o2n    = (const int*)d_in[1];
    const float* Bh     = (const float*)d_in[2];
    const float* Bc     = (const float*)d_in[3];
    const float* Uh     = (const float*)d_in[4];
    const float* Uc     = (const float*)d_in[5];
    const float* lalpha = (const float*)d_in[6];
    const float* bias   = (const float*)d_in[7];
    float* out = (float*)d_out;

    char* ws = (char*)d_ws;
    float* A = (float*)ws;  ws += (size_t)BT_ * RF_ * sizeof(float);
    float* G = (float*)ws;  ws += (size_t)RF_ * RF_ * sizeof(float);
    float* P = (float*)ws;  ws += (size_t)BT_ * RF_ * sizeof(float);
    int* hot_pos = (int*)ws;

    // Independent prep
    k_invmap <<<dim3((V_ + 255) / 256), dim3(256), 0, stream>>>(o2n, hot_pos);
    k_build_a<<<dim3(2, BT_),           dim3(256), 0, stream>>>(tokens, o2n, Uh, Uc, A);
    k_gram   <<<dim3(26 * 26),          dim3(32),  0, stream>>>(Bh, Bc, G);
    // P = A @ G  (encode + decode-projection fused via Gram trick)
    k_project<<<dim3(26, BT_ / 128),    dim3(256), 0, stream>>>(A, G, P);
    // Full-V rank-32 GEMM with permutation + alpha + bias fused (writes d_out once)
    int coldTilesX = (V_ + 16 * CT_PER_WG_ - 1) / (16 * CT_PER_WG_);   // 393
    k_cold   <<<dim3(coldTilesX, BT_ / 128), dim3(256), 0, stream>>>(P, Uc, o2n, bias, lalpha, out);
    // 2% hot columns, rank-384, scatter-overwrite (ordered after k_cold)
    k_hot    <<<dim3(K_ / 16, BT_ / 128),     dim3(256), 0, stream>>>(P, Uh, hot_pos, bias, lalpha, out);
}